// GCN_84456236908760
// MI455X (gfx1250) — compile-verified
//
#include <hip/hip_runtime.h>

typedef __attribute__((ext_vector_type(2))) float    v2f;
typedef __attribute__((ext_vector_type(8))) float    v8f;
typedef __attribute__((ext_vector_type(4))) unsigned v4u;
typedef __attribute__((ext_vector_type(8))) unsigned v8u;

#define D 128            // feature dim (both layers)
#define ROWS 80          // rows per GEMM block (100000 = 80 * 1250)

// ---------------- degree / normalization ----------------

__global__ void k_deg_init(float* __restrict__ deg, int n) {
    int i = blockIdx.x * blockDim.x + threadIdx.x;
    if (i < n) deg[i] = 1.0f;            // self-loop contributes 1 to every node
}

__global__ void k_deg_acc(const int* __restrict__ col, float* __restrict__ deg, int e) {
    int i = blockIdx.x * blockDim.x + threadIdx.x;
    if (i < e) unsafeAtomicAdd(&deg[col[i]], 1.0f);
}

__global__ void k_rsqrt(float* __restrict__ deg, int n) {
    int i = blockIdx.x * blockDim.x + threadIdx.x;
    if (i < n) deg[i] = rsqrtf(deg[i]);  // deg >= 1 always
}

// ---------------- GEMM: C[80 x 128] = A[80 x 128] @ W[128 x 128] ----------------
// One block = 80 output rows, 128 threads (4 waves).
// W (64KB) staged via the Tensor Data Mover (tensor_load_to_lds, TENSORcnt);
// A-tile (40KB) staged with vector loads (fused ReLU for layer 2).
// Each wave computes 5 row-tiles x 2 col-tiles of 16x16 with v_wmma_f32_16x16x4_f32.

__global__ __launch_bounds__(128) void k_gemm(const float* __restrict__ A,
                                              const float* __restrict__ W,
                                              float* __restrict__ C,
                                              int relu, int n) {
    __shared__ float sW[D * D];       // 64 KB
    __shared__ float sA[ROWS * D];    // 40 KB

    const int t    = threadIdx.x;
    const int wave = t >> 5;
    const int lane = t & 31;
    const int row0 = blockIdx.x * ROWS;

    // ---- TDM stage of W: one DMA issued by wave 0 ----
    if (wave == 0) {
        // D# group 0: count=1 | lds_addr | global_addr(57b) | type=2
        unsigned lds_base = (unsigned)(size_t)(&sW[0]);
        unsigned long long ga = (unsigned long long)(size_t)W;
        v4u g0;
        g0[0] = 1u;                                            // count=1, user mode
        g0[1] = lds_base;                                      // lds_addr
        g0[2] = (unsigned)ga;                                  // global_addr[31:0]
        g0[3] = ((unsigned)(ga >> 32) & 0x01FFFFFFu)           // global_addr[56:32]
                | 0x80000000u;                                 // type=2 ("image")
        // D# group 1: data_size=4B; tensor 128x128; tile 128x128; stride0=128
        v8u g1;
        g1[0] = 0x00020000u;      // workgroup_mask=0, data_size=2 (4 bytes)
        g1[1] = 128u << 16;       // tensor_dim0[15:0] in [31:16]
        g1[2] = 128u << 16;       // tensor_dim1[15:0] in [31:16]
        g1[3] = 128u << 16;       // tile_dim0 in [31:16]
        g1[4] = 128u;             // tile_dim1 in [15:0] (tile_dim2 = 0)
        g1[5] = 128u;             // tensor_dim0_stride[31:0]
        g1[6] = 0u;               // stride0 hi / tensor_dim1_stride lo (unused, 2D)
        g1[7] = 0u;
        asm volatile("tensor_load_to_lds %0, %1"
                     :: "s"(g0), "s"(g1)
                     : "memory");
        __builtin_amdgcn_s_wait_tensorcnt(0);
    }

    // ---- cooperative vector load of the 80x128 A tile (fused ReLU) ----
    {
        const float4* A4 = (const float4*)(A + (size_t)row0 * D);
        float4* sA4 = (float4*)sA;
        #pragma unroll
        for (int i = t; i < ROWS * D / 4; i += 128) {
            const int row = i >> 5;                 // 32 float4 per row
            float4 v = make_float4(0.f, 0.f, 0.f, 0.f);
            if (row0 + row < n) v = A4[i];
            if (relu) {
                v.x = fmaxf(v.x, 0.0f); v.y = fmaxf(v.y, 0.0f);
                v.z = fmaxf(v.z, 0.0f); v.w = fmaxf(v.w, 0.0f);
            }
            sA4[i] = v;
        }
    }
    __syncthreads();

    const int h  = lane >> 4;   // half-wave: K pair {k,k+1} vs {k+2,k+3}
    const int r  = lane & 15;   // row (A) / column-in-tile (B, C/D)
    const int c0 = wave * 32;   // this wave's 32-column slab

    v8f acc[5][2] = {};

    for (int k = 0; k < D; k += 4) {
        const int kk = k + 2 * h;
        // B fragments (shared across all 5 row tiles)
        v2f b0, b1;
        b0.x = sW[kk * D + c0 + r];
        b0.y = sW[(kk + 1) * D + c0 + r];
        b1.x = sW[kk * D + c0 + 16 + r];
        b1.y = sW[(kk + 1) * D + c0 + 16 + r];
        #pragma unroll
        for (int rt = 0; rt < 5; ++rt) {
            v2f a;
            a.x = sA[(rt * 16 + r) * D + kk];
            a.y = sA[(rt * 16 + r) * D + kk + 1];
            acc[rt][0] = __builtin_amdgcn_wmma_f32_16x16x4_f32(false, a, false, b0,
                                                              (short)0, acc[rt][0],
                                                              false, false);
            acc[rt][1] = __builtin_amdgcn_wmma_f32_16x16x4_f32(false, a, false, b1,
                                                              (short)0, acc[rt][1],
                                                              false, false);
        }
    }

    // C/D layout: VGPR j holds row M = j + 8*h, column = r (within tile).
    #pragma unroll
    for (int rt = 0; rt < 5; ++rt) {
        #pragma unroll
        for (int j = 0; j < 8; ++j) {
            const int m = row0 + rt * 16 + j + 8 * h;
            if (m < n) {
                C[(size_t)m * D + c0 + r]      = acc[rt][0][j];
                C[(size_t)m * D + c0 + 16 + r] = acc[rt][1][j];
            }
        }
    }
}

// ---------------- self-loop + bias init: out = b + dinv[i]^2 * tmp ----------------

__global__ void k_self_bias(const float* __restrict__ tmp,
                            const float* __restrict__ dinv,
                            const float* __restrict__ b,
                            float* __restrict__ out, int total) {
    int i = blockIdx.x * blockDim.x + threadIdx.x;
    if (i < total) {
        int node = i >> 7;      // / 128
        int d    = i & 127;
        float w = dinv[node];
        out[i] = b[d] + w * w * tmp[i];
    }
}

// ---------------- edge scatter: out[col] += dinv[row]*dinv[col]*tmp[row] ----------------
// One wave (32 lanes) per edge; each lane handles a float4 (512B coalesced gather),
// then 4 f32 global atomic adds (L2-resident for this problem size).

__global__ __launch_bounds__(256) void k_scatter(const float* __restrict__ tmp,
                                                 const float* __restrict__ dinv,
                                                 const int* __restrict__ rows,
                                                 const int* __restrict__ cols,
                                                 float* __restrict__ out, int e) {
    int gid  = blockIdx.x * blockDim.x + threadIdx.x;
    int edge = gid >> 5;
    int lane = gid & 31;
    if (edge >= e) return;

    const int rsrc = rows[edge];
    const int cdst = cols[edge];
    const float w = dinv[rsrc] * dinv[cdst];

    const float4 v = ((const float4*)(tmp + (size_t)rsrc * D))[lane];
    float* o = out + (size_t)cdst * D + lane * 4;
    unsafeAtomicAdd(o + 0, w * v.x);
    unsafeAtomicAdd(o + 1, w * v.y);
    unsafeAtomicAdd(o + 2, w * v.z);
    unsafeAtomicAdd(o + 3, w * v.w);
}

// ---------------- launcher ----------------

extern "C" void kernel_launch(void* const* d_in, const int* in_sizes, int n_in,
                              void* d_out, int out_size, void* d_ws, size_t ws_size,
                              hipStream_t stream) {
    const float* x  = (const float*)d_in[0];
    const int*   ei = (const int*)d_in[1];   // [2, E] flattened: rows then cols
    const float* W1 = (const float*)d_in[2];
    const float* b1 = (const float*)d_in[3];
    const float* W2 = (const float*)d_in[4];
    const float* b2 = (const float*)d_in[5];
    float* out = (float*)d_out;

    const int n = in_sizes[0] / D;      // 100000 nodes
    const int e = in_sizes[1] / 2;      // 1.6M edges
    const int* rows = ei;
    const int* cols = ei + e;

    // Workspace: dinv [n] | tmp [n*D] | h1 [n*D]
    float* dinv = (float*)d_ws;
    float* tmp  = dinv + n;
    float* h1   = tmp + (size_t)n * D;

    const int total = n * D;
    const int gemm_blocks = (n + ROWS - 1) / ROWS;            // 1250
    const int scatter_blocks = (int)(((long long)e * 32 + 255) / 256);

    // Degree / normalization
    k_deg_init<<<(n + 255) / 256, 256, 0, stream>>>(dinv, n);
    k_deg_acc<<<(e + 255) / 256, 256, 0, stream>>>(cols, dinv, e);
    k_rsqrt<<<(n + 255) / 256, 256, 0, stream>>>(dinv, n);

    // Layer 1: h1 = (pre-activation) GCNConv(x, W1, b1)
    k_gemm<<<gemm_blocks, 128, 0, stream>>>(x, W1, tmp, 0, n);
    k_self_bias<<<(total + 255) / 256, 256, 0, stream>>>(tmp, dinv, b1, h1, total);
    k_scatter<<<scatter_blocks, 256, 0, stream>>>(tmp, dinv, rows, cols, h1, e);

    // Layer 2: out = GCNConv(relu(h1), W2, b2)   (ReLU fused into GEMM A-load)
    k_gemm<<<gemm_blocks, 128, 0, stream>>>(h1, W2, tmp, 1, n);
    k_self_bias<<<(total + 255) / 256, 256, 0, stream>>>(tmp, dinv, b2, out, total);
    k_scatter<<<scatter_blocks, 256, 0, stream>>>(tmp, dinv, rows, cols, out, e);
}